// PPOAgent_72018011619397
// MI455X (gfx1250) — compile-verified
//
#include <hip/hip_runtime.h>
#include <math.h>

// ---------------------------------------------------------------------------
// PPO agent forward pass for MI455X (gfx1250, wave32, WMMA).
// Heavy GEMMs run on V_WMMA_F32_16X16X4_F32 (fp32 matrix pipe); well-shaped
// GEMMs feed LDS via global_load_async_to_lds_b128 with double buffering.
// ---------------------------------------------------------------------------

typedef __attribute__((ext_vector_type(2))) float v2f;
typedef __attribute__((ext_vector_type(8))) float v8f;

#define B_    2048
#define D_    32
#define HIST_ 50
#define ADIM_ 60
#define NHEAD_ 4
#define NL_   3
#define DM_   512
#define DFF_  1024
#define OUT_  256
#define FDIM_ 98          // 3*D+2
#define GDIM_ 561         // 2*D+1+D*(D-1)/2
#define CDIM_ 8384        // D*OUT + 128 + 64
#define TOK_  (B_ * D_)   // 65536 tokens

#define BM 64
#define BN 64
#define BK 16
#define LP (BK + 4)       // LDS row pitch: 80B -> 16B-aligned & conflict-free

// =================== fast GEMM: async LDS fill + pipelining ================
// Requirements: M%64==0, N%64==0, K%16==0, lda%4==0, ldw%4==0, 16B-aligned
// base pointers. C = act(A[M,K] @ W[N,K]^T + bias + res).
__global__ __launch_bounds__(256) void gemm_wmma_f32_async(
    const float* __restrict__ A, int lda,
    const float* __restrict__ W, int ldw,
    const float* __restrict__ bias,
    const float* __restrict__ res,
    float* __restrict__ C, int ldc,
    int M, int N, int K,
    int relu, int fold, int foldstride)
{
    __shared__ __align__(16) float As[2][BM][LP];
    __shared__ __align__(16) float Ws[2][BM][LP];

    const int tid  = threadIdx.x;
    const int lane = tid & 31;
    const int wid  = tid >> 5;
    const int wm   = wid >> 1;     // 0..3 : 16-row strip
    const int wn   = wid & 1;      // 0..1 : 32-col strip
    const int half = lane >> 4;
    const int lrow = lane & 15;

    const int m_blk = blockIdx.y * BM;
    const int n_blk = blockIdx.x * BN;

    // one 16B async copy per thread per matrix per stage: 64 rows x 4 segs
    const int lr = tid >> 2;            // 0..63
    const int ls = (tid & 3) << 2;      // 0,4,8,12
    const float* aptr = A + (size_t)(m_blk + lr) * lda + ls;
    const float* wptr = W + (size_t)(n_blk + lr) * ldw + ls;

    v8f acc0 = {};
    v8f acc1 = {};

    auto issue = [&](int k0, int buf) {
        const unsigned la = (unsigned)(size_t)&As[buf][lr][ls];
        const unsigned lw = (unsigned)(size_t)&Ws[buf][lr][ls];
        const float* ga = aptr + k0;
        const float* gw = wptr + k0;
        asm volatile("global_load_async_to_lds_b128 %0, %1, off"
                     :: "v"(la), "v"(ga) : "memory");
        asm volatile("global_load_async_to_lds_b128 %0, %1, off"
                     :: "v"(lw), "v"(gw) : "memory");
    };

    const int nT = K / BK;
    issue(0, 0);

    for (int i = 0; i < nT; ++i) {
        const int buf = i & 1;
        if (i + 1 < nT) {
            issue((i + 1) * BK, buf ^ 1);            // prefetch next tile (DMA)
            asm volatile("s_wait_asynccnt 2" ::: "memory");  // tile i landed
        } else {
            asm volatile("s_wait_asynccnt 0" ::: "memory");
        }
        __syncthreads();

#pragma unroll
        for (int kk = 0; kk < BK; kk += 4) {
            const int ar = wm * 16 + lrow;
            const int ak = kk + 2 * half;
            v2f a;  a.x = As[buf][ar][ak];        a.y = As[buf][ar][ak + 1];
            const int bn0 = wn * 32 + lrow;
            v2f b0; b0.x = Ws[buf][bn0][ak];      b0.y = Ws[buf][bn0][ak + 1];
            v2f b1; b1.x = Ws[buf][bn0 + 16][ak]; b1.y = Ws[buf][bn0 + 16][ak + 1];

            acc0 = __builtin_amdgcn_wmma_f32_16x16x4_f32(
                       false, a, false, b0, (short)0, acc0, false, false);
            acc1 = __builtin_amdgcn_wmma_f32_16x16x4_f32(
                       false, a, false, b1, (short)0, acc1, false, false);
        }
        __syncthreads();   // everyone done with buf before it is re-filled
    }

#pragma unroll
    for (int r = 0; r < 8; ++r) {
        const int row = m_blk + wm * 16 + r + 8 * half;
#pragma unroll
        for (int t = 0; t < 2; ++t) {
            const int col = n_blk + wn * 32 + t * 16 + lrow;
            float v = (t == 0) ? acc0[r] : acc1[r];
            if (bias) v += bias[col];
            if (res)  v += res[(size_t)row * ldc + col];
            if (relu) v = fmaxf(v, 0.0f);
            const size_t off = fold
                ? ((size_t)(row / fold) * ldc + (size_t)(row % fold) * foldstride + col)
                : ((size_t)row * ldc + col);
            C[off] = v;
        }
    }
}

// =================== generic GEMM (bounds-checked K/N tails) ===============
__global__ __launch_bounds__(256) void gemm_wmma_f32(
    const float* __restrict__ A, int lda,
    const float* __restrict__ W, int ldw,
    const float* __restrict__ bias,
    const float* __restrict__ res,
    const float* __restrict__ pos,   // per (row % 32, col) additive, ld = N
    float* __restrict__ C, int ldc,
    int M, int N, int K,
    int relu, int fold, int foldstride)
{
    __shared__ float As[BM][LP];
    __shared__ float Ws[BM][LP];

    const int tid  = threadIdx.x;
    const int lane = tid & 31;
    const int wid  = tid >> 5;
    const int wm   = wid >> 1;
    const int wn   = wid & 1;
    const int half = lane >> 4;
    const int lrow = lane & 15;

    const int m_blk = blockIdx.y * BM;
    const int n_blk = blockIdx.x * BN;

    v8f acc0 = {};
    v8f acc1 = {};

    const int lr = tid >> 2;            // 0..63
    const int lc = (tid & 3) << 2;      // 0,4,8,12
    const int gr = m_blk + lr;          // M divisible by 64 always
    const int gn = n_blk + lr;
    const bool wok = (gn < N);

    for (int k0 = 0; k0 < K; k0 += BK) {
        {
            const float* ap = A + (size_t)gr * lda + k0 + lc;
            const float* wp = W + (size_t)gn * ldw + k0 + lc;
            const bool fullk = (k0 + BK <= K);
            if (fullk) {
#pragma unroll
                for (int c = 0; c < 4; ++c) {
                    As[lr][lc + c] = ap[c];
                    Ws[lr][lc + c] = wok ? wp[c] : 0.0f;
                }
                if (k0 + BK < K) {
                    __builtin_prefetch(ap + BK, 0, 1);
                    if (wok) __builtin_prefetch(wp + BK, 0, 1);
                }
            } else {
#pragma unroll
                for (int c = 0; c < 4; ++c) {
                    const int k = k0 + lc + c;
                    As[lr][lc + c] = (k < K) ? ap[c] : 0.0f;
                    Ws[lr][lc + c] = (k < K && wok) ? wp[c] : 0.0f;
                }
            }
        }
        __syncthreads();

#pragma unroll
        for (int kk = 0; kk < BK; kk += 4) {
            const int ar = wm * 16 + lrow;
            const int ak = kk + 2 * half;
            v2f a;  a.x = As[ar][ak];        a.y = As[ar][ak + 1];
            const int bn0 = wn * 32 + lrow;
            v2f b0; b0.x = Ws[bn0][ak];      b0.y = Ws[bn0][ak + 1];
            v2f b1; b1.x = Ws[bn0 + 16][ak]; b1.y = Ws[bn0 + 16][ak + 1];

            acc0 = __builtin_amdgcn_wmma_f32_16x16x4_f32(
                       false, a, false, b0, (short)0, acc0, false, false);
            acc1 = __builtin_amdgcn_wmma_f32_16x16x4_f32(
                       false, a, false, b1, (short)0, acc1, false, false);
        }
        __syncthreads();
    }

#pragma unroll
    for (int r = 0; r < 8; ++r) {
        const int row = m_blk + wm * 16 + r + 8 * half;
#pragma unroll
        for (int t = 0; t < 2; ++t) {
            const int col = n_blk + wn * 32 + t * 16 + lrow;
            if (col >= N) continue;
            float v = (t == 0) ? acc0[r] : acc1[r];
            if (bias) v += bias[col];
            if (pos)  v += pos[(size_t)(row & (D_ - 1)) * N + col];
            if (res)  v += res[(size_t)row * ldc + col];
            if (relu) v = fmaxf(v, 0.0f);
            const size_t off = fold
                ? ((size_t)(row / fold) * ldc + (size_t)(row % fold) * foldstride + col)
                : ((size_t)row * ldc + col);
            C[off] = v;
        }
    }
}

// ==================== Gram-Schmidt + vertex/global features ================
__global__ __launch_bounds__(32) void gs_features_kernel(
    const float* __restrict__ basis, float* __restrict__ vf, float* __restrict__ gf)
{
    const int b = blockIdx.x;
    const int lane = threadIdx.x;

    __shared__ float Bs[D_][D_ + 1];
    __shared__ float Gs[D_][D_ + 1];
    __shared__ float Ms[D_];
    __shared__ float bns[D_];
    __shared__ float gns[D_];

    for (int j = 0; j < D_; ++j)
        Bs[j][lane] = basis[((size_t)b * D_ + j) * D_ + lane];
    __syncthreads();

    for (int j = 0; j < D_; ++j) {
        float dot = 0.0f, sq = 0.0f;
        if (lane < j) {
#pragma unroll 4
            for (int d = 0; d < D_; ++d) {
                const float g = Gs[lane][d];
                dot += g * Bs[j][d];
                sq  += g * g;
            }
        }
        const float mu = (lane < j) ? dot / (sq > 0.0f ? sq : 1.0f) : 0.0f;
        Ms[lane] = mu;
        __syncthreads();

        float gj = Bs[j][lane];
        for (int k = 0; k < j; ++k) gj -= Ms[k] * Gs[k][lane];
        Gs[j][lane] = gj;

        const float bv = Bs[j][lane];
        float s1 = bv * bv, s2 = gj * gj;
#pragma unroll
        for (int m = 16; m > 0; m >>= 1) {
            s1 += __shfl_xor(s1, m, 32);
            s2 += __shfl_xor(s2, m, 32);
        }
        const float bnv = sqrtf(s1);
        const float gnv = sqrtf(s2);
        if (lane == 0) { bns[j] = bnv; gns[j] = gnv; }

        const size_t vb = ((size_t)b * D_ + j) * FDIM_;
        vf[vb + lane]        = bv;
        vf[vb + D_ + lane]   = gj;
        if (lane == 0) { vf[vb + 64] = bnv; vf[vb + 65] = gnv; }
        vf[vb + 66 + lane]   = mu;
        __syncthreads();
    }

    float detp = 1.0f, bnp = 1.0f;
    for (int k = 0; k < D_; ++k) { detp *= gns[k]; bnp *= bns[k]; }
    const float odef = bnp / (detp + 1e-10f);

    const size_t gb = (size_t)b * GDIM_;
    if (lane == 0) gf[gb] = odef;

    for (int p = lane; p < 496; p += 32) {
        int pp = p, i = 0;
        while (pp >= D_ - 1 - i) { pp -= D_ - 1 - i; ++i; }
        const int jj = i + 1 + pp;
        float d = 0.0f;
#pragma unroll 4
        for (int c = 0; c < D_; ++c) d += Bs[i][c] * Bs[jj][c];
        gf[gb + 1 + p] = d / ((bns[i] + 1e-10f) * (bns[jj] + 1e-10f));
    }
    gf[gb + 497 + lane] = bns[lane];
    gf[gb + 529 + lane] = gns[lane];
}

// ============================ attention (S=32, dh=128) =====================
__global__ __launch_bounds__(128) void attention_kernel(
    const float* __restrict__ qkv, float* __restrict__ o)
{
    const int bh = blockIdx.x;
    const int b  = bh / NHEAD_;
    const int h  = bh % NHEAD_;
    const int tid  = threadIdx.x;
    const int lane = tid & 31;
    const int wid  = tid >> 5;

    __shared__ float Qs[D_][129];
    __shared__ float Ks[D_][129];
    __shared__ float Vs[D_][129];
    __shared__ float Sc[D_][D_ + 1];

    for (int e = tid; e < D_ * 128; e += 128) {
        const int s = e >> 7, d = e & 127;
        const size_t base = ((size_t)(b * D_ + s)) * (3 * DM_) + h * 128;
        Qs[s][d] = qkv[base + d];
        Ks[s][d] = qkv[base + DM_ + d];
        Vs[s][d] = qkv[base + 2 * DM_ + d];
    }
    __syncthreads();

    const float scale = 0.08838834764831845f;  // 128^-0.5
    for (int p = tid; p < D_ * D_; p += 128) {
        const int i = p >> 5, j = p & 31;
        float acc = 0.0f;
#pragma unroll 4
        for (int d = 0; d < 128; ++d) acc += Qs[i][d] * Ks[j][d];
        Sc[i][j] = acc * scale;
    }
    __syncthreads();

    for (int i = wid; i < D_; i += 4) {
        float v = Sc[i][lane];
        float mx = v;
#pragma unroll
        for (int m = 16; m > 0; m >>= 1) mx = fmaxf(mx, __shfl_xor(mx, m, 32));
        const float e = __expf(v - mx);
        float sm = e;
#pragma unroll
        for (int m = 16; m > 0; m >>= 1) sm += __shfl_xor(sm, m, 32);
        Sc[i][lane] = e / sm;
    }
    __syncthreads();

    const int d = tid;
    for (int i = 0; i < D_; ++i) {
        float acc = 0.0f;
#pragma unroll 4
        for (int j = 0; j < D_; ++j) acc += Sc[i][j] * Vs[j][d];
        o[((size_t)(b * D_ + i)) * DM_ + h * 128 + d] = acc;
    }
}

// ================================ layernorm ================================
__global__ __launch_bounds__(256) void layernorm_kernel(
    float* __restrict__ x, const float* __restrict__ g, const float* __restrict__ bt)
{
    const int row = blockIdx.x;
    const int tid = threadIdx.x;
    const int lane = tid & 31;
    const int wid  = tid >> 5;

    __shared__ float rs[8], rq[8];
    __shared__ float s_mean, s_rstd;

    float* xr = x + (size_t)row * DM_;
    const float v0 = xr[tid];
    const float v1 = xr[tid + 256];
    float sum = v0 + v1;
    float sq  = v0 * v0 + v1 * v1;
#pragma unroll
    for (int m = 16; m > 0; m >>= 1) {
        sum += __shfl_xor(sum, m, 32);
        sq  += __shfl_xor(sq,  m, 32);
    }
    if (lane == 0) { rs[wid] = sum; rq[wid] = sq; }
    __syncthreads();
    if (tid == 0) {
        float S = 0.0f, Q = 0.0f;
#pragma unroll
        for (int w = 0; w < 8; ++w) { S += rs[w]; Q += rq[w]; }
        const float mean = S * (1.0f / DM_);
        const float var  = Q * (1.0f / DM_) - mean * mean;
        s_mean = mean;
        s_rstd = rsqrtf(var + 1e-5f);
    }
    __syncthreads();
    const float mean = s_mean, rstd = s_rstd;
    xr[tid]       = (v0 - mean) * rstd * g[tid]       + bt[tid];
    xr[tid + 256] = (v1 - mean) * rstd * g[tid + 256] + bt[tid + 256];
}

// ========================= action-history embedding gather =================
__global__ __launch_bounds__(256) void ae_gather_kernel(
    const int* __restrict__ ah, const float* __restrict__ act_emb, float* __restrict__ ae0)
{
    const size_t e = (size_t)blockIdx.x * blockDim.x + threadIdx.x;
    if (e >= (size_t)B_ * HIST_ * 32) return;
    const int b = (int)(e / (HIST_ * 32));
    const int r = (int)(e % (HIST_ * 32));
    const int hh = r >> 5, c = r & 31;
    int a = ah[b * HIST_ + hh];
    if (a == -1) a = ADIM_;
    ae0[e] = act_emb[a * 32 + c];
}

// =============================== softmax (N=60) ============================
__global__ __launch_bounds__(32) void softmax60_kernel(
    const float* __restrict__ logits, float* __restrict__ out)
{
    const int b = blockIdx.x;
    const int lane = threadIdx.x;
    const float v0 = logits[(size_t)b * ADIM_ + lane];
    const float v1 = (lane < ADIM_ - 32) ? logits[(size_t)b * ADIM_ + 32 + lane] : -3.0e38f;
    float mx = fmaxf(v0, v1);
#pragma unroll
    for (int m = 16; m > 0; m >>= 1) mx = fmaxf(mx, __shfl_xor(mx, m, 32));
    const float e0 = __expf(v0 - mx);
    const float e1 = (lane < ADIM_ - 32) ? __expf(v1 - mx) : 0.0f;
    float s = e0 + e1;
#pragma unroll
    for (int m = 16; m > 0; m >>= 1) s += __shfl_xor(s, m, 32);
    const float inv = 1.0f / s;
    out[(size_t)b * ADIM_ + lane] = e0 * inv;
    if (lane < ADIM_ - 32) out[(size_t)b * ADIM_ + 32 + lane] = e1 * inv;
}

// ============================== value head (N=1) ===========================
__global__ __launch_bounds__(32) void value_kernel(
    const float* __restrict__ h2, const float* __restrict__ w,
    const float* __restrict__ bv, float* __restrict__ out)
{
    const int b = blockIdx.x;
    const int lane = threadIdx.x;
    float acc = 0.0f;
    for (int c = lane; c < DM_; c += 32) acc += h2[(size_t)b * DM_ + c] * w[c];
#pragma unroll
    for (int m = 16; m > 0; m >>= 1) acc += __shfl_xor(acc, m, 32);
    if (lane == 0) out[b] = acc + bv[0];
}

// ================================ launcher =================================
extern "C" void kernel_launch(void* const* d_in, const int* in_sizes, int n_in,
                              void* d_out, int out_size, void* d_ws, size_t ws_size,
                              hipStream_t stream) {
    (void)in_sizes; (void)n_in; (void)out_size; (void)ws_size;

    const float* basis   = (const float*)d_in[0];
    const int*   ah      = (const int*)  d_in[1];
    const float* emb_W   = (const float*)d_in[2];
    const float* emb_b   = (const float*)d_in[3];
    const float* pos     = (const float*)d_in[4];
    const float* qkv_W   = (const float*)d_in[5];
    const float* qkv_b   = (const float*)d_in[6];
    const float* out_W   = (const float*)d_in[7];
    const float* out_b   = (const float*)d_in[8];
    const float* ln1_g   = (const float*)d_in[9];
    const float* ln1_b   = (const float*)d_in[10];
    const float* ln2_g   = (const float*)d_in[11];
    const float* ln2_b   = (const float*)d_in[12];
    const float* ff1_W   = (const float*)d_in[13];
    const float* ff1_b   = (const float*)d_in[14];
    const float* ff2_W   = (const float*)d_in[15];
    const float* ff2_b   = (const float*)d_in[16];
    const float* proj_W  = (const float*)d_in[17];
    const float* proj_b  = (const float*)d_in[18];
    const float* glob1_W = (const float*)d_in[19];
    const float* glob1_b = (const float*)d_in[20];
    const float* glob2_W = (const float*)d_in[21];
    const float* glob2_b = (const float*)d_in[22];
    const float* act_emb = (const float*)d_in[23];
    const float* act1_W  = (const float*)d_in[24];
    const float* act1_b  = (const float*)d_in[25];
    const float* act2_W  = (const float*)d_in[26];
    const float* act2_b  = (const float*)d_in[27];
    const float* a1_W    = (const float*)d_in[28];
    const float* a1_b    = (const float*)d_in[29];
    const float* a2_W    = (const float*)d_in[30];
    const float* a2_b    = (const float*)d_in[31];
    const float* a3_W    = (const float*)d_in[32];
    const float* a3_b    = (const float*)d_in[33];
    const float* c1_W    = (const float*)d_in[34];
    const float* c1_b    = (const float*)d_in[35];
    const float* c2_W    = (const float*)d_in[36];
    const float* c2_b    = (const float*)d_in[37];
    const float* c3_W    = (const float*)d_in[38];
    const float* c3_b    = (const float*)d_in[39];

    float* ws = (float*)d_ws;
    size_t off = 0;
    float* VF   = ws + off; off += (size_t)TOK_ * FDIM_;   // 65536 x 98
    float* X0   = ws + off; off += (size_t)TOK_ * DM_;     // 65536 x 512
    float* X1   = ws + off; off += (size_t)TOK_ * DM_;     // 65536 x 512
    float* QKV  = ws + off; off += (size_t)TOK_ * 3 * DM_; // 65536 x 1536
    float* OH   = ws + off; off += (size_t)TOK_ * DFF_;    // attn-out aliases ff-hidden
    float* COMB = ws + off; off += (size_t)B_ * CDIM_;     // 2048 x 8384
    float* GF   = ws + off; off += (size_t)B_ * GDIM_;     // 2048 x 561
    float* AE0  = ws + off; off += (size_t)B_ * HIST_ * 32;// 2048 x 1600
    float* TG1  = ws + off; off += (size_t)B_ * 256;
    float* TA1  = ws + off; off += (size_t)B_ * 128;
    float* H1   = ws + off; off += (size_t)B_ * DM_;
    float* H2   = ws + off; off += (size_t)B_ * DM_;
    float* LOG  = ws + off; off += (size_t)B_ * 64;

    auto gemm = [&](const float* A, int lda, const float* W, int ldw,
                    const float* bias, const float* res, const float* posp,
                    float* C, int ldc, int M, int N, int K,
                    int relu, int fold, int foldstride) {
        dim3 grid((N + BN - 1) / BN, M / BM);
        const bool fast = (posp == nullptr) && (K % BK == 0) && (N % BN == 0) &&
                          (lda % 4 == 0) && (ldw % 4 == 0);
        if (fast) {
            gemm_wmma_f32_async<<<grid, 256, 0, stream>>>(
                A, lda, W, ldw, bias, res, C, ldc, M, N, K, relu, fold, foldstride);
        } else {
            gemm_wmma_f32<<<grid, 256, 0, stream>>>(
                A, lda, W, ldw, bias, res, posp, C, ldc, M, N, K, relu, fold, foldstride);
        }
    };

    // feature extraction
    gs_features_kernel<<<B_, 32, 0, stream>>>(basis, VF, GF);
    {
        const size_t n = (size_t)B_ * HIST_ * 32;
        ae_gather_kernel<<<(int)((n + 255) / 256), 256, 0, stream>>>(ah, act_emb, AE0);
    }

    // embedding: x = vf @ emb_W^T + emb_b + pos (K=98 -> generic kernel)
    gemm(VF, FDIM_, emb_W, FDIM_, emb_b, nullptr, pos, X0, DM_, TOK_, DM_, FDIM_, 0, 0, 0);

    // transformer layers (post-norm, fused residuals in GEMM epilogue)
    for (int l = 0; l < NL_; ++l) {
        gemm(X0, DM_, qkv_W + (size_t)l * 3 * DM_ * DM_, DM_, qkv_b + (size_t)l * 3 * DM_,
             nullptr, nullptr, QKV, 3 * DM_, TOK_, 3 * DM_, DM_, 0, 0, 0);
        attention_kernel<<<B_ * NHEAD_, 128, 0, stream>>>(QKV, OH);
        gemm(OH, DM_, out_W + (size_t)l * DM_ * DM_, DM_, out_b + (size_t)l * DM_,
             X0, nullptr, X1, DM_, TOK_, DM_, DM_, 0, 0, 0);
        layernorm_kernel<<<TOK_, 256, 0, stream>>>(X1, ln1_g + (size_t)l * DM_, ln1_b + (size_t)l * DM_);
        gemm(X1, DM_, ff1_W + (size_t)l * DFF_ * DM_, DM_, ff1_b + (size_t)l * DFF_,
             nullptr, nullptr, OH, DFF_, TOK_, DFF_, DM_, 1, 0, 0);
        gemm(OH, DFF_, ff2_W + (size_t)l * DM_ * DFF_, DFF_, ff2_b + (size_t)l * DM_,
             X1, nullptr, X0, DM_, TOK_, DM_, DFF_, 0, 0, 0);
        layernorm_kernel<<<TOK_, 256, 0, stream>>>(X0, ln2_g + (size_t)l * DM_, ln2_b + (size_t)l * DM_);
    }

    // token projection -> flat [B, D*256] written straight into comb[:, :8192]
    gemm(X0, DM_, proj_W, DM_, proj_b, nullptr, nullptr, COMB, CDIM_,
         TOK_, OUT_, DM_, 0, D_, OUT_);
    // global-feature MLP -> comb[:, 8192:8320]  (K=561 -> generic kernel)
    gemm(GF, GDIM_, glob1_W, GDIM_, glob1_b, nullptr, nullptr, TG1, 256, B_, 256, GDIM_, 1, 0, 0);
    gemm(TG1, 256, glob2_W, 256, glob2_b, nullptr, nullptr, COMB + D_ * OUT_, CDIM_, B_, 128, 256, 0, 0, 0);
    // action-history MLP -> comb[:, 8320:8384]
    gemm(AE0, HIST_ * 32, act1_W, HIST_ * 32, act1_b, nullptr, nullptr, TA1, 128, B_, 128, HIST_ * 32, 1, 0, 0);
    gemm(TA1, 128, act2_W, 128, act2_b, nullptr, nullptr, COMB + D_ * OUT_ + 128, CDIM_, B_, 64, 128, 0, 0, 0);

    float* probs = (float*)d_out;
    float* value = (float*)d_out + (size_t)B_ * ADIM_;

    // actor head
    gemm(COMB, CDIM_, a1_W, CDIM_, a1_b, nullptr, nullptr, H1, DM_, B_, DM_, CDIM_, 1, 0, 0);
    gemm(H1, DM_, a2_W, DM_, a2_b, nullptr, nullptr, H2, DM_, B_, DM_, DM_, 1, 0, 0);
    gemm(H2, DM_, a3_W, DM_, a3_b, nullptr, nullptr, LOG, ADIM_, B_, ADIM_, DM_, 0, 0, 0);  // N=60 -> generic
    softmax60_kernel<<<B_, 32, 0, stream>>>(LOG, probs);

    // critic head
    gemm(COMB, CDIM_, c1_W, CDIM_, c1_b, nullptr, nullptr, H1, DM_, B_, DM_, CDIM_, 1, 0, 0);
    gemm(H1, DM_, c2_W, DM_, c2_b, nullptr, nullptr, H2, DM_, B_, DM_, DM_, 1, 0, 0);
    value_kernel<<<B_, 32, 0, stream>>>(H2, c3_W, c3_b, value);
}